// TwoLayerReducedGammaModel_18674517803550
// MI455X (gfx1250) — compile-verified
//
#include <hip/hip_runtime.h>
#include <hip/hip_bf16.h>

typedef __attribute__((ext_vector_type(2))) float v2f;
typedef __attribute__((ext_vector_type(4))) float v4f;
typedef __attribute__((ext_vector_type(8))) float v8f;

// ---------------------------------------------------------------------------
// Kernel 1: xty[d] = sum_p X[p,d] * y[p]
// ---------------------------------------------------------------------------
__global__ void k_xty(const float* __restrict__ X, const float* __restrict__ y,
                      float* __restrict__ xty, int D) {
  int d = blockIdx.x * 256 + threadIdx.x;
  int p0 = blockIdx.y * 128;
  float s = 0.f;
  for (int p = p0; p < p0 + 128; ++p)
    s = fmaf(X[(size_t)p * D + d], y[p], s);
  atomicAdd(&xty[d], s);
}

// ---------------------------------------------------------------------------
// Kernel 2: Gram matrix G = X^T X (unnormalized), f32 WMMA 16x16x4.
// Block = 4 waves, each wave computes a 32x32 region (2x2 WMMA tiles);
// block covers 64x64 of the 1024x1024 output.
// A[m,k] = X[p0+k, d1b+m], B[k,n] = X[p0+k, d2b+n].
// 32-bit operand layout: lanes 0-15 -> (K=k0 in VGPR0, K=k0+1 in VGPR1),
// lanes 16-31 -> (K=k0+2, K=k0+3), M/N = lane&15.
// ---------------------------------------------------------------------------
__global__ __launch_bounds__(128) void k_gram(const float* __restrict__ X,
                                              float* __restrict__ G, int P, int D) {
  const int lane = threadIdx.x & 31;
  const int wave = threadIdx.x >> 5;
  const int h = lane >> 4, l = lane & 15;
  const int d1b = blockIdx.x * 64 + (wave >> 1) * 32;
  const int d2b = blockIdx.y * 64 + (wave & 1) * 32;

  v8f c00 = {}, c01 = {}, c10 = {}, c11 = {};
  for (int p0 = 0; p0 < P; p0 += 4) {
    const float* r0 = X + (size_t)(p0 + 2 * h) * D;
    const float* r1 = r0 + D;
    v2f a0, a1, b0, b1;
    a0.x = r0[d1b + l];       a0.y = r1[d1b + l];
    a1.x = r0[d1b + 16 + l];  a1.y = r1[d1b + 16 + l];
    b0.x = r0[d2b + l];       b0.y = r1[d2b + l];
    b1.x = r0[d2b + 16 + l];  b1.y = r1[d2b + 16 + l];
    c00 = __builtin_amdgcn_wmma_f32_16x16x4_f32(false, a0, false, b0, (short)0, c00, false, false);
    c01 = __builtin_amdgcn_wmma_f32_16x16x4_f32(false, a0, false, b1, (short)0, c01, false, false);
    c10 = __builtin_amdgcn_wmma_f32_16x16x4_f32(false, a1, false, b0, (short)0, c10, false, false);
    c11 = __builtin_amdgcn_wmma_f32_16x16x4_f32(false, a1, false, b1, (short)0, c11, false, false);
  }
  // C layout: lane n holds N = n&15; VGPR j holds M = j + 8*(n>=16)
  #pragma unroll
  for (int j = 0; j < 8; ++j) {
    int m0 = d1b + 8 * h + j;
    G[(size_t)m0 * D + d2b + l]        = c00[j];
    G[(size_t)m0 * D + d2b + 16 + l]   = c01[j];
    G[(size_t)(m0 + 16) * D + d2b + l]      = c10[j];
    G[(size_t)(m0 + 16) * D + d2b + 16 + l] = c11[j];
  }
}

// ---------------------------------------------------------------------------
// Kernel 3: persistent power-series kernel.
// acc = sum_{l=0}^{L-1} w_l ; w_{l+1} = w_l - (gamma/(L*P)) * G * w_l ; w_0 = xty
// 64 blocks x 256 threads (16 threads per row), software device-wide barrier.
// Writes rep = gamma*acc and norm2 = |rep|^2.
// ---------------------------------------------------------------------------
__global__ __launch_bounds__(256) void k_power(const float* __restrict__ G,
                                               const float* __restrict__ xty,
                                               float* __restrict__ w0, float* __restrict__ w1,
                                               float* __restrict__ rep, float* __restrict__ norm2,
                                               unsigned* __restrict__ ctr,
                                               const float* __restrict__ gamma_p,
                                               const int* __restrict__ L_p,
                                               int P, int D, int nblocks) {
  __shared__ float wlds[1024];
  const int sub = threadIdx.x & 15;
  const int r = blockIdx.x * 16 + (threadIdx.x >> 4);
  const float gamma = *gamma_p;
  const int L = *L_p;
  const float c = gamma / ((float)L * (float)P);
  float acc = 0.f;

  for (int l = 0; l < L; ++l) {
    const float* wc = (l == 0) ? xty : ((l & 1) ? w0 : w1);
    float* wn = (l & 1) ? w1 : w0;
    for (int i = threadIdx.x; i < D; i += 256) wlds[i] = wc[i];
    __syncthreads();

    float s = 0.f;
    const float* grow = G + (size_t)r * D;
    for (int j = 0; j < D / 16; ++j) {
      int col = sub + 16 * j;
      s = fmaf(grow[col], wlds[col], s);
    }
    s += __shfl_xor(s, 8, 16);
    s += __shfl_xor(s, 4, 16);
    s += __shfl_xor(s, 2, 16);
    s += __shfl_xor(s, 1, 16);
    if (sub == 0) {
      float wr = wlds[r];
      acc += wr;
      wn[r] = wr - c * s;
    }
    // device-wide barrier (monotonic counter; ctr zeroed each launch)
    __threadfence();
    __syncthreads();
    if (threadIdx.x == 0) {
      atomicAdd(ctr, 1u);
      unsigned target = (unsigned)nblocks * (unsigned)(l + 1);
      while (__hip_atomic_load(ctr, __ATOMIC_RELAXED, __HIP_MEMORY_SCOPE_AGENT) < target)
        __builtin_amdgcn_s_sleep(8);
    }
    __syncthreads();
    __threadfence();
  }
  if (sub == 0) {
    float rv = gamma * acc;
    rep[r] = rv;
    atomicAdd(norm2, rv * rv);
  }
}

// ---------------------------------------------------------------------------
// Kernel 4: build extended B matrix (col-of-output major, i.e. same layout as W):
//  Bt[m][d] = W[m][d] * rep_n[d]          (m < M)
//  Bt[M][d] = rep[d] / (L*P)              (linear term column, shift trick)
//  Bt[m][d] = 0                           (padding cols)
// a_ext / b1_ext extended accordingly (a_ext[M]=1, b1_ext[M]=1 -> subtract 1 in bias).
// ---------------------------------------------------------------------------
__global__ void k_prep(const float* __restrict__ W, const float* __restrict__ a,
                       const float* __restrict__ b1, const float* __restrict__ rep,
                       const float* __restrict__ norm2, const int* __restrict__ L_p,
                       float* __restrict__ Bt, float* __restrict__ aext,
                       float* __restrict__ b1ext, int P, int D, int M, int NCOL) {
  int idx = blockIdx.x * blockDim.x + threadIdx.x;
  int total = NCOL * D;
  if (idx < total) {
    int d = idx & (D - 1);
    int m = idx >> 10;            // D == 1024
    float v;
    if (m < M) {
      float inv = 1.0f / (sqrtf(*norm2) + 1e-8f);
      v = W[(size_t)m * D + d] * rep[d] * inv;
    } else if (m == M) {
      int L = *L_p;
      v = rep[d] / ((float)L * (float)P);
    } else {
      v = 0.f;
    }
    Bt[idx] = v;
  }
  if (idx < NCOL) {
    aext[idx]  = (idx < M) ? a[idx]  : (idx == M ? 1.0f : 0.0f);
    b1ext[idx] = (idx < M) ? b1[idx] : (idx == M ? 1.0f : 0.0f);
  }
}

// ---------------------------------------------------------------------------
// Kernel 5: initialize out[k] = b2 - 1  (absorbs the relu shift trick)
// ---------------------------------------------------------------------------
__global__ void k_init_out(float* __restrict__ out, const float* __restrict__ b2, int K) {
  int i = blockIdx.x * blockDim.x + threadIdx.x;
  if (i < K) out[i] = *b2 - 1.0f;
}

// ---------------------------------------------------------------------------
// Kernel 6: fused GEMM + ReLU + weighted reduce.
// out[k] += sum_m a_ext[m] * relu( X_star[k,:] . Bt[m,:] + b1_ext[m] )
// Workgroup = 4 waves, 64 rows of X_star staged in LDS (256KB, XOR swizzled),
// 16-column B tile staged per iteration (64KB).  f32 WMMA 16x16x4, K=1024.
// ---------------------------------------------------------------------------
__global__ __launch_bounds__(128) void k_main(const float* __restrict__ Xs,
                                              const float* __restrict__ Bt,
                                              const float* __restrict__ aext,
                                              const float* __restrict__ b1ext,
                                              float* __restrict__ out, int D, int NT) {
  extern __shared__ float smem[];
  float* As = smem;               // [64][1024], k ^= ((row&15)<<2) swizzle
  float* Bs = smem + 64 * 1024;   // [16][1024], k ^= (c<<2) swizzle

  const int tid = threadIdx.x;
  const int lane = tid & 31, wave = tid >> 5;
  const int h = lane >> 4, l = lane & 15;
  const long R0 = (long)blockIdx.x * 64;

  // stage 64-row X_star slab (16384 float4, 128 per thread)
  #pragma unroll 4
  for (int i = 0; i < 128; ++i) {
    int f4 = tid + 128 * i;
    int row = f4 >> 8;                 // 256 float4 per row
    int kf = (f4 & 255) << 2;
    v4f v = *(const v4f*)(Xs + (R0 + row) * D + kf);
    *(v4f*)(As + row * 1024 + (kf ^ ((row & 15) << 2))) = v;
  }

  float rowacc[8];
  #pragma unroll
  for (int j = 0; j < 8; ++j) rowacc[j] = 0.f;

  for (int ct = 0; ct < NT; ++ct) {
    __syncthreads();                   // protect Bs overwrite
    #pragma unroll 4
    for (int i = 0; i < 32; ++i) {     // stage 16-column B tile
      int f4 = tid + 128 * i;
      int c = f4 >> 8;
      int kf = (f4 & 255) << 2;
      v4f v = *(const v4f*)(Bt + (size_t)(ct * 16 + c) * D + kf);
      *(v4f*)(Bs + c * 1024 + (kf ^ (c << 2))) = v;
    }
    __syncthreads();

    v8f C = {};
    const int arow = wave * 16 + l;
    const float* ap = As + arow * 1024;
    const int asw = (arow & 15) << 2;
    const float* bp = Bs + l * 1024;
    const int bsw = l << 2;
    #pragma unroll 8
    for (int k0 = 0; k0 < 1024; k0 += 4) {
      int ka = k0 + 2 * h;
      v2f av = *(const v2f*)(ap + (ka ^ asw));
      v2f bv = *(const v2f*)(bp + (ka ^ bsw));
      C = __builtin_amdgcn_wmma_f32_16x16x4_f32(false, av, false, bv, (short)0, C, false, false);
    }

    int col = ct * 16 + l;
    float avw = aext[col], bvw = b1ext[col];
    #pragma unroll
    for (int j = 0; j < 8; ++j) {
      float t = fmaxf(C[j] + bvw, 0.0f) * avw;
      t += __shfl_xor(t, 8, 16);
      t += __shfl_xor(t, 4, 16);
      t += __shfl_xor(t, 2, 16);
      t += __shfl_xor(t, 1, 16);
      if (l == 0) rowacc[j] += t;      // lanes 0 and 16: rows j and j+8
    }
  }

  if (l == 0) {
    #pragma unroll
    for (int j = 0; j < 8; ++j) {
      long row = R0 + wave * 16 + h * 8 + j;
      out[row] += rowacc[j];           // single writer per row; out pre-filled
    }
  }
}

// ---------------------------------------------------------------------------
extern "C" void kernel_launch(void* const* d_in, const int* in_sizes, int n_in,
                              void* d_out, int out_size, void* d_ws, size_t ws_size,
                              hipStream_t stream) {
  const float* X     = (const float*)d_in[0];
  const float* y     = (const float*)d_in[1];
  const float* Xs    = (const float*)d_in[2];
  const float* gamma = (const float*)d_in[3];
  const float* W     = (const float*)d_in[4];
  const float* a     = (const float*)d_in[5];
  const float* b1    = (const float*)d_in[6];
  const float* b2    = (const float*)d_in[7];
  const int*   Lp    = (const int*)d_in[8];
  float* out = (float*)d_out;

  const int P = in_sizes[1];           // 8192
  const int D = in_sizes[0] / P;       // 1024
  const int K = in_sizes[2] / D;       // 131072
  const int M = in_sizes[4] / D;       // 512
  const int NCOL = 528, NT = 33;       // 512 MLP cols + 1 linear col + pad

  // workspace layout (floats)
  float* ws    = (float*)d_ws;
  float* xty   = ws + 0;               // 1024
  float* norm2 = ws + 1024;            // 1
  unsigned* ctr = (unsigned*)(ws + 1025);
  float* w0    = ws + 1280;            // 1024
  float* w1    = ws + 2304;            // 1024
  float* rep   = ws + 3328;            // 1024
  float* aext  = ws + 4352;            // 528
  float* b1ext = ws + 4880;            // 528
  float* G     = ws + 5632;            // 1024*1024
  float* Bt    = ws + 5632 + 1024 * 1024; // 528*1024

  hipMemsetAsync(d_ws, 0, 5120, stream);   // zero xty, norm2, ctr

  k_xty<<<dim3(D / 256, P / 128), 256, 0, stream>>>(X, y, xty, D);
  k_gram<<<dim3(D / 64, D / 64), 128, 0, stream>>>(X, G, P, D);
  k_power<<<D / 16, 256, 0, stream>>>(G, xty, w0, w1, rep, norm2, ctr,
                                      gamma, Lp, P, D, D / 16);
  int total = NCOL * D;
  k_prep<<<(total + 255) / 256, 256, 0, stream>>>(W, a, b1, rep, norm2, Lp,
                                                  Bt, aext, b1ext, P, D, M, NCOL);
  k_init_out<<<(K + 255) / 256, 256, 0, stream>>>(out, b2, K);

  hipFuncSetAttribute((const void*)k_main,
                      hipFuncAttributeMaxDynamicSharedMemorySize, 327680);
  k_main<<<K / 64, 128, 327680, stream>>>(Xs, Bt, aext, b1ext, out, D, NT);
}